// LayoutDiffusionLoss_30167850287357
// MI455X (gfx1250) — compile-verified
//
#include <hip/hip_runtime.h>
#include <hip/hip_bf16.h>

// LayoutDiffusionLoss for MI455X (gfx1250, wave32).
// B=64, N=256, D=6. Three kernels:
//  k1: elementwise losses + per-batch sums (one block per batch)
//  k2: pairwise losses staged through LDS (one block per batch)
//  k3: single-wave finalize; 64->1 partial reductions via V_WMMA_F32_16X16X4_F32

#define EPSF     1e-8f
#define GOLDENF  1.618f

static constexpr int BB = 64;    // batch
static constexpr int NN = 256;   // elements per batch
// ws float layout:
//  [0,64)    A1   = sum(area)      per batch
//  [64,128)  A2   = sum(area^2)    per batch
//  [128,192) SCX  = sum(cx)        per batch
//  [192,256) SCY  = sum(cy)        per batch
//  [256,320) VC   = sum(mask)      per batch
//  [320, 320+10*64) block partials: acc*64 + block
//   acc: 0 noise, 1 posmse, 2 iou, 3 layermse, 4 confmse, 5 golden,
//        6 rank, 7 overlap, 8 align  (9 unused spare)
static constexpr int PART = 320;

typedef __attribute__((ext_vector_type(2))) float v2f;
typedef __attribute__((ext_vector_type(8))) float v8f;

#if defined(__has_builtin)
#if __has_builtin(__builtin_amdgcn_wmma_f32_16x16x4_f32)
#define HAVE_WMMA_F32X4 1
#endif
#endif

__device__ __forceinline__ float wave_sum32(float v) {
#pragma unroll
  for (int o = 16; o; o >>= 1) v += __shfl_xor(v, o, 32);
  return v;  // same value in all 32 lanes
}

// Deterministic block reduction (blockDim.x == 256 => 8 waves). Result valid in thread 0.
__device__ __forceinline__ float block_reduce256(float v, float* smem8) {
  int lane = threadIdx.x & 31;
  int w    = threadIdx.x >> 5;
  v = wave_sum32(v);
  if (lane == 0) smem8[w] = v;
  __syncthreads();
  float r = 0.f;
  if (threadIdx.x == 0) {
#pragma unroll
    for (int i = 0; i < 8; ++i) r += smem8[i];
  }
  __syncthreads();
  return r;
}

// Sum 64 f32 values using one V_WMMA_F32_16X16X4_F32.
// B = all-ones => every element of D is a row-sum of A; summing all 256 D
// elements gives 16 * sum(A) independent of the exact VGPR lane layout.
__device__ __forceinline__ float reduce64(const float* __restrict__ p, int lane) {
#if defined(HAVE_WMMA_F32X4)
  v2f a;  a.x = p[2 * lane]; a.y = p[2 * lane + 1];
  v2f bo; bo.x = 1.0f;       bo.y = 1.0f;
  v8f c = {0.f, 0.f, 0.f, 0.f, 0.f, 0.f, 0.f, 0.f};
  v8f d = __builtin_amdgcn_wmma_f32_16x16x4_f32(
      /*neg_a=*/false, a, /*neg_b=*/false, bo,
      /*c_mod=*/(short)0, c, /*reuse_a=*/false, /*reuse_b=*/false);
  float s = 0.f;
#pragma unroll
  for (int i = 0; i < 8; ++i) s += d[i];
  return wave_sum32(s) * 0.0625f;  // /16
#else
  float s = p[2 * lane] + p[2 * lane + 1];
  return wave_sum32(s);
#endif
}

// ---------------- Kernel 1: elementwise + per-batch sums ----------------
__global__ __launch_bounds__(256) void ldl_k1(
    const float* __restrict__ pn, const float* __restrict__ tn,
    const float* __restrict__ pl, const float* __restrict__ tl,
    const float* __restrict__ mk, float* __restrict__ ws) {
  __shared__ float smem8[8];
  const int b = blockIdx.x;
  const int n = threadIdx.x;           // blockDim.x == NN
  const int idx = b * NN + n;
  const float m = mk[idx];
  const float* pnr = pn + (size_t)idx * 6;
  const float* tnr = tn + (size_t)idx * 6;
  const float* plr = pl + (size_t)idx * 6;
  const float* tlr = tl + (size_t)idx * 6;

  // SmoothL1 over masked noise (all 6 dims)
  float s_noise = 0.f;
#pragma unroll
  for (int d = 0; d < 6; ++d) {
    float dd = fabsf(pnr[d] * m - tnr[d] * m);
    s_noise += (dd < 1.f) ? 0.5f * dd * dd : dd - 0.5f;
  }

  const float p0 = plr[0] * m, p1 = plr[1] * m, p2 = plr[2] * m;
  const float p3 = plr[3] * m, p4 = plr[4] * m, p5 = plr[5] * m;
  const float t0 = tlr[0] * m, t1 = tlr[1] * m, t2 = tlr[2] * m;
  const float t3 = tlr[3] * m, t4 = tlr[4] * m, t5 = tlr[5] * m;

  // position MSE over first 4 dims
  const float s_pos = (p0 - t0) * (p0 - t0) + (p1 - t1) * (p1 - t1) +
                      (p2 - t2) * (p2 - t2) + (p3 - t3) * (p3 - t3);

  // IoU loss term (1 - iou)
  const float iw = fmaxf(fminf(p0 + p2, t0 + t2) - fmaxf(p0, t0), 0.f);
  const float ih = fmaxf(fminf(p1 + p3, t1 + t3) - fmaxf(p1, t1), 0.f);
  const float inter = iw * ih;
  const float iou = inter / (p2 * p3 + t2 * t3 - inter + EPSF);
  const float s_iou = 1.f - iou;

  const float s_lm = (p4 - t4) * (p4 - t4);
  const float s_cm = (p5 - t5) * (p5 - t5);
  const float s_gold = fabsf(p2 / (p3 + EPSF) - GOLDENF);

  // aesthetic per-batch sums (mask re-applied, faithful to reference)
  const float area = p2 * p3 * m;
  const float cx = (p0 + 0.5f * p2) * m;
  const float cy = (p1 + 0.5f * p3) * m;

  float r;
  r = block_reduce256(s_noise, smem8); if (threadIdx.x == 0) ws[PART + 0 * 64 + b] = r;
  r = block_reduce256(s_pos,   smem8); if (threadIdx.x == 0) ws[PART + 1 * 64 + b] = r;
  r = block_reduce256(s_iou,   smem8); if (threadIdx.x == 0) ws[PART + 2 * 64 + b] = r;
  r = block_reduce256(s_lm,    smem8); if (threadIdx.x == 0) ws[PART + 3 * 64 + b] = r;
  r = block_reduce256(s_cm,    smem8); if (threadIdx.x == 0) ws[PART + 4 * 64 + b] = r;
  r = block_reduce256(s_gold,  smem8); if (threadIdx.x == 0) ws[PART + 5 * 64 + b] = r;
  r = block_reduce256(area,        smem8); if (threadIdx.x == 0) ws[0   + b] = r;
  r = block_reduce256(area * area, smem8); if (threadIdx.x == 0) ws[64  + b] = r;
  r = block_reduce256(cx,          smem8); if (threadIdx.x == 0) ws[128 + b] = r;
  r = block_reduce256(cy,          smem8); if (threadIdx.x == 0) ws[192 + b] = r;
  r = block_reduce256(m,           smem8); if (threadIdx.x == 0) ws[256 + b] = r;
}

// ---------------- Kernel 2: pairwise losses (LDS-staged) ----------------
__global__ __launch_bounds__(256) void ldl_k2(
    const float* __restrict__ pl, const float* __restrict__ tl,
    const float* __restrict__ mk, float* __restrict__ ws) {
  __shared__ float sx1[NN], sy1[NN], sx2[NN], sy2[NN], scx[NN], scy[NN];
  __shared__ float sar[NN], sp4[NN], st4[NN], smk[NN];
  __shared__ float smem8[8];
  const int b = blockIdx.x;

  for (int n = threadIdx.x; n < NN; n += 256) {
    const int idx = b * NN + n;
    const float m = mk[idx];
    const float* plr = pl + (size_t)idx * 6;
    const float* tlr = tl + (size_t)idx * 6;
    const float x1 = plr[0] * m, y1 = plr[1] * m;
    const float w  = plr[2] * m, h  = plr[3] * m;
    sx1[n] = x1;            sy1[n] = y1;
    sx2[n] = x1 + w;        sy2[n] = y1 + h;
    scx[n] = x1 + 0.5f * w; scy[n] = y1 + 0.5f * h;
    sar[n] = w * h;
    sp4[n] = plr[4] * m;    st4[n] = tlr[4] * m;
    smk[n] = m;
  }
  __syncthreads();

  float s_rank = 0.f, s_ovl = 0.f, s_al = 0.f;
  for (int i = 0; i < NN - 1; ++i) {
    const float x1i = sx1[i], y1i = sy1[i], x2i = sx2[i], y2i = sy2[i];
    const float cxi = scx[i], cyi = scy[i], ari = sar[i];
    const float p4i = sp4[i], t4i = st4[i], mi = smk[i];
    for (int j = i + 1 + (int)threadIdx.x; j < NN; j += 256) {
      // ranking BCE (no pair_valid in reference): d = p[j]-p[i], y = (t[i]<t[j])
      const float d  = sp4[j] - p4i;
      const float lt = log1pf(expf(-fabsf(d)));     // stable softplus tail
      const float sp_pos = fmaxf(d, 0.f) + lt;      // softplus(d)
      const float sp_neg = fmaxf(-d, 0.f) + lt;     // softplus(-d)
      const float y = (t4i < st4[j]) ? 1.f : 0.f;
      s_rank += y * fminf(sp_neg, 100.f) + (1.f - y) * fminf(sp_pos, 100.f);

      const float pv = mi * smk[j];

      // overlap ratio
      const float iw = fmaxf(fminf(x2i, sx2[j]) - fmaxf(x1i, sx1[j]), 0.f);
      const float ih = fmaxf(fminf(y2i, sy2[j]) - fmaxf(y1i, sy1[j]), 0.f);
      const float mina = fminf(ari, sar[j]);
      s_ovl += (iw * ih) / (mina + EPSF) * pv;

      // alignment: 6 edge types, penalty dist/5 when dist < 5
      float pen = 0.f;
      float e;
      e = fabsf(x1i - sx1[j]); pen += (e < 5.f) ? e * 0.2f : 0.f;
      e = fabsf(x2i - sx2[j]); pen += (e < 5.f) ? e * 0.2f : 0.f;
      e = fabsf(y1i - sy1[j]); pen += (e < 5.f) ? e * 0.2f : 0.f;
      e = fabsf(y2i - sy2[j]); pen += (e < 5.f) ? e * 0.2f : 0.f;
      e = fabsf(cxi - scx[j]); pen += (e < 5.f) ? e * 0.2f : 0.f;
      e = fabsf(cyi - scy[j]); pen += (e < 5.f) ? e * 0.2f : 0.f;
      s_al += pen * pv;
    }
  }

  float r;
  r = block_reduce256(s_rank, smem8); if (threadIdx.x == 0) ws[PART + 6 * 64 + b] = r;
  r = block_reduce256(s_ovl,  smem8); if (threadIdx.x == 0) ws[PART + 7 * 64 + b] = r;
  r = block_reduce256(s_al,   smem8); if (threadIdx.x == 0) ws[PART + 8 * 64 + b] = r;
}

// ---------------- Kernel 3: single-wave finalize (WMMA reductions) ----------------
__global__ __launch_bounds__(32) void ldl_k3(const float* __restrict__ ws,
                                             float* __restrict__ out) {
  const int lane = threadIdx.x;  // 32 threads, EXEC all ones for WMMA

  const float S_noise = reduce64(ws + PART + 0 * 64, lane);
  const float S_pos   = reduce64(ws + PART + 1 * 64, lane);
  const float S_iou   = reduce64(ws + PART + 2 * 64, lane);
  const float S_lm    = reduce64(ws + PART + 3 * 64, lane);
  const float S_cm    = reduce64(ws + PART + 4 * 64, lane);
  const float S_gold  = reduce64(ws + PART + 5 * 64, lane);
  const float S_rank  = reduce64(ws + PART + 6 * 64, lane);
  const float S_ovl   = reduce64(ws + PART + 7 * 64, lane);
  const float S_al    = reduce64(ws + PART + 8 * 64, lane);
  const float S_A1    = reduce64(ws + 0, lane);   // total area sum over all B*N

  // size harmony: mean_b std(areas_b, ddof=1) / (mean(areas) + eps)
  float sstd = 0.f;
#pragma unroll
  for (int b = lane; b < BB; b += 32) {
    const float a1 = ws[b], a2 = ws[64 + b];
    const float var = (a2 - a1 * a1 * (1.f / 256.f)) * (1.f / 255.f);
    sstd += sqrtf(fmaxf(var, 0.f));
  }
  sstd = wave_sum32(sstd);
  const float mean_area = S_A1 * (1.f / (BB * (float)NN));
  const float size_harm = (sstd * (1.f / BB)) / (mean_area + EPSF);

  // balance with faithful [B]/[B,1] -> [B,B] broadcast
  float bal = 0.f;
  for (int t = lane; t < BB * BB; t += 32) {
    const int a  = t >> 6;      // row: selects valid_count
    const int b2 = t & 63;      // col: selects center sums
    const float vc = ws[256 + a] + EPSF;
    const float ox = ws[128 + b2] / vc;
    const float oy = ws[192 + b2] / vc;
    bal += (fabsf(ox - 512.f) + fabsf(oy - 512.f)) * (1.f / 512.f);
  }
  bal = wave_sum32(bal) * (1.f / (BB * (float)BB));

  const float invBN     = 1.f / (BB * (float)NN);            // 1/16384
  const float invPairs  = 1.f / (BB * 32640.f);               // B * N(N-1)/2
  const float noise_loss = S_noise * (1.f / (BB * (float)NN * 6.f));
  const float position_loss = S_pos * (1.f / (BB * (float)NN * 4.f)) + 0.5f * (S_iou * invBN);
  const float ranking = S_rank * invPairs;
  const float layer_loss = S_lm * invBN + 0.3f * ranking;
  const float conf_loss  = S_cm * invBN;
  const float golden     = S_gold * invBN;
  const float aesthetic  = 0.1f * golden + 0.2f * size_harm + 0.15f * bal;
  const float overlap    = S_ovl * invPairs;
  const float alignment  = S_al * (invPairs * (1.f / 6.f));

  const float total = noise_loss + position_loss + 0.5f * layer_loss +
                      0.3f * conf_loss + 0.2f * aesthetic +
                      0.8f * overlap + 0.4f * alignment;
  if (lane == 0) out[0] = total;
}

extern "C" void kernel_launch(void* const* d_in, const int* in_sizes, int n_in,
                              void* d_out, int out_size, void* d_ws, size_t ws_size,
                              hipStream_t stream) {
  (void)in_sizes; (void)n_in; (void)out_size; (void)ws_size;
  const float* pn = (const float*)d_in[0];
  const float* tn = (const float*)d_in[1];
  const float* pl = (const float*)d_in[2];
  const float* tl = (const float*)d_in[3];
  const float* mk = (const float*)d_in[4];
  float* ws  = (float*)d_ws;   // needs 320 + 10*64 = 960 floats (3.75 KB)
  float* out = (float*)d_out;

  ldl_k1<<<BB, 256, 0, stream>>>(pn, tn, pl, tl, mk, ws);
  ldl_k2<<<BB, 256, 0, stream>>>(pl, tl, mk, ws);
  ldl_k3<<<1, 32, 0, stream>>>(ws, out);
}